// YoloV3Loss_66133906424309
// MI455X (gfx1250) — compile-verified
//
#include <hip/hip_runtime.h>
#include <hip/hip_bf16.h>

// ---------------- problem constants ----------------
#define NP   22743          // anchors per image = (76^2+38^2+19^2)*3
#define LVL0 17328          // 76*76*3
#define LVL1 21660          // + 38*38*3
#define NT   20             // targets per image
#define NC   80             // classes
#define NB   16             // batch

__device__ __constant__ float c_anch[18] = {
    10,13, 16,30, 33,23, 30,61, 62,45, 59,119, 116,90, 156,198, 373,326};

// ---------------- helpers ----------------
__device__ __forceinline__ float sigmoidf(float x) { return 1.0f / (1.0f + expf(-x)); }
// softplus(x) = log(1+e^x), numerically stable
__device__ __forceinline__ float softplusf(float x) {
    return fmaxf(x, 0.0f) + log1pf(expf(-fabsf(x)));
}

// row pointer into the virtual concatenated predicts[b, i, 0:85]
__device__ __forceinline__ const float* pred_row(const float* __restrict__ f0,
                                                 const float* __restrict__ f1,
                                                 const float* __restrict__ f2,
                                                 int b, int i) {
    if (i < LVL0) return f0 + ((size_t)b * LVL0 + i) * 85;
    if (i < LVL1) return f1 + ((size_t)b * 4332 + (i - LVL0)) * 85;
    return f2 + ((size_t)b * 1083 + (i - LVL1)) * 85;
}

// ---------------- WMMA wave reduction (CDNA5) ----------------
// Total sum of one f32 value per lane across a wave32 using
// V_WMMA_F32_16X16X4_F32: A = ones(16x4), lane values placed in B's first
// VGPR (they occupy 32 distinct B slots whatever the K striping), second
// VGPR = 0.  D[m][n] = colsum[n] replicated over m; d[0] per lane holds
// colsum[lane%16]; 4 xor-adds inside each 16-lane half give the total.
typedef float v2f __attribute__((ext_vector_type(2)));
typedef float v8f __attribute__((ext_vector_type(8)));

__device__ __forceinline__ float wave_reduce_wmma(float v) {
    v2f a; a[0] = 1.0f; a[1] = 1.0f;
    v2f bb; bb[0] = v; bb[1] = 0.0f;
    v8f c = {};
    v8f d = __builtin_amdgcn_wmma_f32_16x16x4_f32(
        /*neg_a=*/false, a, /*neg_b=*/false, bb,
        /*c_mod=*/(short)0, c, /*reuse_a=*/false, /*reuse_b=*/false);
    float r = d[0];
    r += __shfl_xor(r, 1, 32);
    r += __shfl_xor(r, 2, 32);
    r += __shfl_xor(r, 4, 32);
    r += __shfl_xor(r, 8, 32);
    return r;   // every lane holds the wave total
}

// ================= kernel 1: per-target losses (320 targets total) =========
// ws layout (floats): [0..15]=lbox  [16..31]=lobj  [32..47]=lcls
//                     [48..63]=noobj_sum  [64..79]=noobj_cnt
//                     (int*)(ws+80): pos_idx[16][20]
__global__ void yolo_targets_kernel(const float* __restrict__ f0,
                                    const float* __restrict__ f1,
                                    const float* __restrict__ f2,
                                    const float* __restrict__ tbox,
                                    const int*   __restrict__ tlab,
                                    const float* __restrict__ tsco,
                                    float* __restrict__ ws) {
    __shared__ float s_lbox[NT], s_lobj[NT];
    __shared__ int   s_pos[NT];
    __shared__ float s_red[256];
    const int b = blockIdx.x, tid = threadIdx.x;

    if (tid < NT) {
        const float* box = tbox + (b * NT + tid) * 4;
        const float cx = box[0], cy = box[1], tw = box[2], th = box[3];
        // wh-IoU argmax over 9 anchors (first max wins)
        int best = 0; float bestv = -1.0f;
        const float ta = tw * th;
        for (int a = 0; a < 9; ++a) {
            float aw = c_anch[2 * a], ah = c_anch[2 * a + 1];
            float inter = fminf(tw, aw) * fminf(th, ah);
            float iou = inter / (ta + aw * ah - inter);
            if (iou > bestv) { bestv = iou; best = a; }
        }
        const int   bidx  = best % 3, fi = best / 3;
        const float aw    = c_anch[2 * best], ah = c_anch[2 * best + 1];
        const float scale = (fi == 0) ? 8.0f : ((fi == 1) ? 16.0f : 32.0f);
        const int   h     = (fi == 0) ? 76 : ((fi == 1) ? 38 : 19);
        const int   start = (fi == 0) ? 0 : ((fi == 1) ? LVL0 : LVL1);

        float sx = cx / scale, sy = cy / scale;
        float fx = sx - floorf(sx); if (fx == 0.0f) fx = 1.0f;
        float fy = sy - floorf(sy); if (fy == 0.0f) fy = 1.0f;
        const int tlx = (int)(sx - fx), tly = (int)(sy - fy);
        const float gtw = logf(tw / aw), gth = logf(th / ah);
        const int pidx = start + (tlx * h + tly) * 3 + bidx;
        s_pos[tid] = pidx;

        const float* p = pred_row(f0, f1, f2, b, pidx);
        const float p0 = p[0], p1 = p[1], p2 = p[2], p3 = p[3], p4 = p[4];
        const float d0 = sigmoidf(p0) - fx, d1 = sigmoidf(p1) - fy;
        const float d2 = p2 - gtw,          d3 = p3 - gth;
        const float w  = 2.0f - gtw * gth;
        s_lbox[tid] = w * (d0 * d0 + d1 * d1 + d2 * d2 + d3 * d3);
        const float t = tsco[b * NT + tid];
        // bce(sigmoid(p4), t) with log clips at -100
        s_lobj[tid] = t * fminf(softplusf(-p4), 100.0f)
                    + (1.0f - t) * fminf(softplusf(p4), 100.0f);
    }
    __syncthreads();

    // class BCE: 20*80 terms spread across the block
    float acc = 0.0f;
    for (int idx = tid; idx < NT * NC; idx += blockDim.x) {
        const int t = idx / NC, c = idx % NC;
        const float* p = pred_row(f0, f1, f2, b, s_pos[t]);
        const float x = p[5 + c];
        const int lab = tlab[b * NT + t];
        acc += (c == lab) ? fminf(softplusf(-x), 100.0f)
                          : fminf(softplusf(x), 100.0f);
    }
    s_red[tid] = acc;
    __syncthreads();
    for (int s = 128; s > 0; s >>= 1) {
        if (tid < s) s_red[tid] += s_red[tid + s];
        __syncthreads();
    }

    if (tid == 0) {
        float lb = 0.0f, lo = 0.0f;
        for (int t = 0; t < NT; ++t) { lb += s_lbox[t]; lo += s_lobj[t]; }
        lb /= (float)(NT * 4);
        if (isinf(lb)) lb = 0.0f;
        ws[b]      = lb;
        ws[16 + b] = lo / (float)NT;
        ws[32 + b] = s_red[0] / (float)(NT * NC);
    }
    int* gpos = (int*)(ws + 80);
    if (tid < NT) gpos[b * NT + tid] = s_pos[tid];
}

// ================= kernel 2: no-obj loss over all anchors ==================
#define APT 4   // anchors per thread
__global__ void yolo_noobj_kernel(const float* __restrict__ f0,
                                  const float* __restrict__ f1,
                                  const float* __restrict__ f2,
                                  const float* __restrict__ tbox,
                                  float* __restrict__ ws) {
    __shared__ float4 s_tb[NT];
    __shared__ float  s_ta[NT];
    __shared__ int    s_pos[NT];
    const int b = blockIdx.y, tid = threadIdx.x;

    if (tid < NT) {
        const float* box = tbox + (b * NT + tid) * 4;
        float4 v = make_float4(box[0], box[1], box[2], box[3]);
        s_tb[tid] = v;
        s_ta[tid] = v.z * v.w;
        s_pos[tid] = ((const int*)(ws + 80))[b * NT + tid];
    }
    __syncthreads();

    float ssum = 0.0f, scnt = 0.0f;
    const int base = blockIdx.x * (blockDim.x * APT);
    #pragma unroll
    for (int k = 0; k < APT; ++k) {
        const int i = base + k * 256 + tid;
        if (i < NP) {
            const int   fi    = (i < LVL0) ? 0 : ((i < LVL1) ? 1 : 2);
            const int   start = (fi == 0) ? 0 : ((fi == 1) ? LVL0 : LVL1);
            const int   h     = (fi == 0) ? 76 : ((fi == 1) ? 38 : 19);
            const float scale = (fi == 0) ? 8.0f : ((fi == 1) ? 16.0f : 32.0f);
            const int local = i - start;
            const int cell  = local / 3, bidx = local - cell * 3;
            const int cx    = cell / h,  cy   = cell - cx * h;

            const float* p = pred_row(f0, f1, f2, b, i);
            if (i + 256 < NP)   // hide the strided 340B-row latency
                __builtin_prefetch(pred_row(f0, f1, f2, b, i + 256), 0, 0);
            const float p0 = p[0], p1 = p[1], p2 = p[2], p3 = p[3], p4 = p[4];

            const float aw = c_anch[(fi * 3 + bidx) * 2];
            const float ah = c_anch[(fi * 3 + bidx) * 2 + 1];
            const float px = ((float)cx + sigmoidf(p0)) * scale;
            const float py = ((float)cy + sigmoidf(p1)) * scale;
            const float pw = aw * expf(p2), ph = ah * expf(p3);
            const float pa = pw * ph;
            const float plx = px - 0.5f * pw, ply = py - 0.5f * ph;
            const float prx = px + 0.5f * pw, pry = py + 0.5f * ph;

            float miou = 0.0f;   // IoUs are non-negative
            #pragma unroll 5
            for (int t = 0; t < NT; ++t) {
                const float4 tb = s_tb[t];
                const float lx = fmaxf(plx, tb.x - 0.5f * tb.z);
                const float ly = fmaxf(ply, tb.y - 0.5f * tb.w);
                const float rx = fminf(prx, tb.x + 0.5f * tb.z);
                const float ry = fminf(pry, tb.y + 0.5f * tb.w);
                const float iw = fmaxf(rx - lx, 0.0f);
                const float ih = fmaxf(ry - ly, 0.0f);
                const float inter = iw * ih;
                miou = fmaxf(miou, inter / (pa + s_ta[t] - inter));
            }
            bool ispos = false;
            #pragma unroll
            for (int t = 0; t < NT; ++t) ispos |= (s_pos[t] == i);
            if (miou < 0.5f && !ispos) {
                ssum += fminf(softplusf(p4), 100.0f);   // -clip(log1p(-sig),-100)
                scnt += 1.0f;
            }
        }
    }

    // CDNA5 WMMA wave reduction (all 32 lanes converged here)
    const float wsum = wave_reduce_wmma(ssum);
    const float wcnt = wave_reduce_wmma(scnt);

    __shared__ float s_ws[8], s_wc[8];
    const int wid = tid >> 5, lane = tid & 31;
    if (lane == 0) { s_ws[wid] = wsum; s_wc[wid] = wcnt; }
    __syncthreads();
    if (tid == 0) {
        float a = 0.0f, c = 0.0f;
        for (int w = 0; w < 8; ++w) { a += s_ws[w]; c += s_wc[w]; }
        atomicAdd(&ws[48 + b], a);
        atomicAdd(&ws[64 + b], c);
    }
}

// ================= kernel 3: finalize 4 outputs ============================
__global__ void yolo_final_kernel(const float* __restrict__ ws,
                                  float* __restrict__ out) {
    if (threadIdx.x == 0 && blockIdx.x == 0) {
        float lb = 0.0f, lo = 0.0f, lc = 0.0f, ln = 0.0f;
        for (int b = 0; b < NB; ++b) {
            lb += ws[b];
            lo += ws[16 + b];
            lc += ws[32 + b];
            ln += ws[48 + b] / fmaxf(ws[64 + b], 1.0f);
        }
        out[0] = lb / 16.0f;            // L_COORD * lbox.mean()
        out[1] = lc / 16.0f;            // L_CLS   * lcls.mean()
        out[2] = lo / 16.0f;            // L_OBJ   * lobj.mean()
        out[3] = 0.5f * (ln / 16.0f);   // L_NOOBJ * lnoobj.mean()
    }
}

// ================= launch ==================================================
extern "C" void kernel_launch(void* const* d_in, const int* in_sizes, int n_in,
                              void* d_out, int out_size, void* d_ws, size_t ws_size,
                              hipStream_t stream) {
    const float* f0   = (const float*)d_in[0];
    const float* f1   = (const float*)d_in[1];
    const float* f2   = (const float*)d_in[2];
    const float* tbox = (const float*)d_in[3];
    const int*   tlab = (const int*)  d_in[4];
    const float* tsco = (const float*)d_in[5];
    float* ws  = (float*)d_ws;
    float* out = (float*)d_out;

    // zero accumulators + pos_idx region: 80 floats + 320 ints = 1600 bytes
    hipMemsetAsync(d_ws, 0, 1600, stream);

    yolo_targets_kernel<<<NB, 256, 0, stream>>>(f0, f1, f2, tbox, tlab, tsco, ws);

    const int gx = (NP + 256 * APT - 1) / (256 * APT);   // 23
    yolo_noobj_kernel<<<dim3(gx, NB), 256, 0, stream>>>(f0, f1, f2, tbox, ws);

    yolo_final_kernel<<<1, 32, 0, stream>>>(ws, out);
}